// BiMambaMixerModel_40561671143731
// MI455X (gfx1250) — compile-verified
//
#include <hip/hip_runtime.h>
#include <hip/hip_bf16.h>
#include <math.h>

typedef __attribute__((ext_vector_type(2))) float v2f;
typedef __attribute__((ext_vector_type(8))) float v8f;

static constexpr int L   = 2048;
static constexpr int D   = 768;
static constexpr int DI  = 1536;   // d_inner
static constexpr int DS  = 16;     // d_state
static constexpr int DTR = 48;     // dt_rank
static constexpr int XP  = 80;     // dt_rank + 2*d_state
static constexpr int XZW = 2 * DI; // 3072

#define WMMA_F32(a, b, c) \
    __builtin_amdgcn_wmma_f32_16x16x4_f32(false, (a), false, (b), (short)0, (c), false, false)

__device__ __forceinline__ void store_tile(float* __restrict__ C, int N,
                                           int row0, int col0, int lane,
                                           const v8f& acc)
{
    float* Cr = C + (size_t)(row0 + ((lane >> 4) << 3)) * N + col0 + (lane & 15);
    #pragma unroll
    for (int i = 0; i < 8; ++i) Cr[(size_t)i * N] = acc[i];
}

// ---------------------------------------------------------------------------
// f32 GEMM via V_WMMA_F32_16X16X4_F32, 4x2 register-blocked:
//   C(M,N) = A(M,K;lda) * W(N,K)^T
// Each wave owns a 64x32 output block (8 accumulators); 6 fragment loads feed
// 8 independent WMMAs per K-step. Loop versioned on the wave-uniform hasN1 so
// no EXEC manipulation occurs inside the hot loop (EXEC stays all-1s).
// Requires M%64==0, N%16==0, K%4==0 (true for all uses here).
// ---------------------------------------------------------------------------
__global__ __launch_bounds__(256) void gemm_f32_wmma(
    const float* __restrict__ A, int lda,
    const float* __restrict__ W,      // (N,K) row-major
    float* __restrict__ C,            // (M,N) row-major
    int M, int N, int K)
{
    const int wave = threadIdx.x >> 5;
    const int lane = threadIdx.x & 31;
    const int tilesN  = N >> 4;
    const int groupsN = (tilesN + 1) >> 1;
    const int groupsM = M >> 6;               // 4 M-tiles per wave
    const int gid = blockIdx.x * 8 + wave;
    if (gid >= groupsM * groupsN) return;     // wave-uniform: EXEC stays all-1s
    const int gm  = gid / groupsN;
    const int gn  = gid % groupsN;
    const int tn0 = gn * 2;
    const bool hasN1 = (tn0 + 1) < tilesN;    // wave-uniform

    const int r  = lane & 15;                 // M index (A) / N index (B)
    const int kh = (lane >> 4) << 1;          // 0 or 2 (K sub-offset)

    const float* A0 = A + (size_t)(gm * 64 + r) * lda + kh;
    const float* A1 = A0 + (size_t)16 * lda;
    const float* A2 = A0 + (size_t)32 * lda;
    const float* A3 = A0 + (size_t)48 * lda;
    const float* B0 = W + (size_t)(tn0 * 16 + r) * K + kh;

    v8f acc00 = {}, acc10 = {}, acc20 = {}, acc30 = {};
    v8f acc01 = {}, acc11 = {}, acc21 = {}, acc31 = {};

    if (hasN1) {
        const float* B1 = W + (size_t)((tn0 + 1) * 16 + r) * K + kh;
        #pragma unroll 2
        for (int k = 0; k < K; k += 4) {
            v2f a0 = *(const v2f*)(A0 + k);
            v2f a1 = *(const v2f*)(A1 + k);
            v2f a2 = *(const v2f*)(A2 + k);
            v2f a3 = *(const v2f*)(A3 + k);
            v2f b0 = *(const v2f*)(B0 + k);
            v2f b1 = *(const v2f*)(B1 + k);
            acc00 = WMMA_F32(a0, b0, acc00);
            acc10 = WMMA_F32(a1, b0, acc10);
            acc20 = WMMA_F32(a2, b0, acc20);
            acc30 = WMMA_F32(a3, b0, acc30);
            acc01 = WMMA_F32(a0, b1, acc01);
            acc11 = WMMA_F32(a1, b1, acc11);
            acc21 = WMMA_F32(a2, b1, acc21);
            acc31 = WMMA_F32(a3, b1, acc31);
        }
    } else {
        #pragma unroll 2
        for (int k = 0; k < K; k += 4) {
            v2f a0 = *(const v2f*)(A0 + k);
            v2f a1 = *(const v2f*)(A1 + k);
            v2f a2 = *(const v2f*)(A2 + k);
            v2f a3 = *(const v2f*)(A3 + k);
            v2f b0 = *(const v2f*)(B0 + k);
            acc00 = WMMA_F32(a0, b0, acc00);
            acc10 = WMMA_F32(a1, b0, acc10);
            acc20 = WMMA_F32(a2, b0, acc20);
            acc30 = WMMA_F32(a3, b0, acc30);
        }
    }

    const int m0 = gm * 64, c0 = tn0 * 16;
    store_tile(C, N, m0,      c0, lane, acc00);
    store_tile(C, N, m0 + 16, c0, lane, acc10);
    store_tile(C, N, m0 + 32, c0, lane, acc20);
    store_tile(C, N, m0 + 48, c0, lane, acc30);
    if (hasN1) {
        store_tile(C, N, m0,      c0 + 16, lane, acc01);
        store_tile(C, N, m0 + 16, c0 + 16, lane, acc11);
        store_tile(C, N, m0 + 32, c0 + 16, lane, acc21);
        store_tile(C, N, m0 + 48, c0 + 16, lane, acc31);
    }
}

// ---------------------------------------------------------------------------
// Embedding gather
// ---------------------------------------------------------------------------
__global__ void embed_kernel(const int* __restrict__ ids,
                             const float* __restrict__ emb,
                             float* __restrict__ hidden)
{
    int idx = blockIdx.x * blockDim.x + threadIdx.x;
    if (idx >= L * D) return;
    int l = idx / D, c = idx % D;
    hidden[idx] = emb[(size_t)ids[l] * D + c];
}

// residual += hidden
__global__ void add_inplace_kernel(float* __restrict__ acc,
                                   const float* __restrict__ v, int n)
{
    int idx = blockIdx.x * blockDim.x + threadIdx.x;
    if (idx < n) acc[idx] += v[idx];
}

// hidden[l] = o_fwd[l] + o_rev[L-1-l]   (un-reverse the reverse direction)
__global__ void combine_kernel(const float* __restrict__ of,
                               const float* __restrict__ orr,
                               float* __restrict__ hidden)
{
    int idx = blockIdx.x * blockDim.x + threadIdx.x;
    if (idx >= L * D) return;
    int l = idx / D, c = idx % D;
    hidden[idx] = of[idx] + orr[(size_t)(L - 1 - l) * D + c];
}

// ---------------------------------------------------------------------------
// LayerNorm: one block per row of D=768
// ---------------------------------------------------------------------------
__global__ __launch_bounds__(256) void layernorm_kernel(
    const float* __restrict__ in, const float* __restrict__ w,
    const float* __restrict__ b, float* __restrict__ out)
{
    __shared__ float s1[256], s2[256];
    const int row = blockIdx.x;
    const float* x = in + (size_t)row * D;
    float a = 0.f, q = 0.f;
    for (int c = threadIdx.x; c < D; c += 256) {
        float v = x[c];
        a += v; q += v * v;
    }
    s1[threadIdx.x] = a; s2[threadIdx.x] = q;
    __syncthreads();
    for (int off = 128; off > 0; off >>= 1) {
        if (threadIdx.x < off) {
            s1[threadIdx.x] += s1[threadIdx.x + off];
            s2[threadIdx.x] += s2[threadIdx.x + off];
        }
        __syncthreads();
    }
    const float mu  = s1[0] * (1.f / D);
    const float var = s2[0] * (1.f / D) - mu * mu;
    const float inv = rsqrtf(var + 1e-5f);
    for (int c = threadIdx.x; c < D; c += 256)
        out[(size_t)row * D + c] = (x[c] - mu) * inv * w[c] + b[c];
}

// ---------------------------------------------------------------------------
// Depthwise causal conv (4 taps) + bias + SiLU; direction reversal folded in.
// ---------------------------------------------------------------------------
__global__ void conv_silu_kernel(const float* __restrict__ xz,
                                 const float* __restrict__ cw,  // (DI,4)
                                 const float* __restrict__ cb,  // (DI)
                                 float* __restrict__ xc, int dir)
{
    int idx = blockIdx.x * blockDim.x + threadIdx.x;
    if (idx >= L * DI) return;
    int t = idx / DI, c = idx % DI;
    float s = cb[c];
    #pragma unroll
    for (int k = 0; k < 4; ++k) {
        int tt = t + k - 3;
        if (tt >= 0) {
            int lsrc = dir ? (L - 1 - tt) : tt;
            s += xz[(size_t)lsrc * XZW + c] * cw[c * 4 + k];
        }
    }
    xc[idx] = s / (1.f + __expf(-s)); // silu
}

// dt = softplus(dt_raw + bias)  in place
__global__ void bias_softplus_kernel(float* __restrict__ dt,
                                     const float* __restrict__ b)
{
    int idx = blockIdx.x * blockDim.x + threadIdx.x;
    if (idx >= L * DI) return;
    float v = dt[idx] + b[idx % DI];
    dt[idx] = (v > 20.f) ? v : log1pf(__expf(v));
}

// ---------------------------------------------------------------------------
// Selective scan: thread = channel, register state h[16], B/C via LDS.
// Output already combines skip (D*x) and SiLU(z) gate.
// ---------------------------------------------------------------------------
__global__ __launch_bounds__(256) void scan_kernel(
    const float* __restrict__ dt,    // (L,DI) post-softplus
    const float* __restrict__ xc,    // (L,DI) post conv+silu
    const float* __restrict__ dbl,   // (L,XP): [..,48:64]=B, [..,64:80]=C
    const float* __restrict__ a_log, // (DI,16)
    const float* __restrict__ dvec,  // (DI)
    const float* __restrict__ xz,    // (L,XZW): z = cols [DI, 2*DI)
    float* __restrict__ y,           // (L,DI)
    int dir)
{
    __shared__ float sBC[32];
    const int ch = blockIdx.x * blockDim.x + threadIdx.x;   // < DI exactly
    float A[DS], h[DS];
    #pragma unroll
    for (int n = 0; n < DS; ++n) {
        A[n] = -__expf(a_log[(size_t)ch * DS + n]);
        h[n] = 0.f;
    }
    const float dv = dvec[ch];
    for (int t = 0; t < L; ++t) {
        if (threadIdx.x < 32) sBC[threadIdx.x] = dbl[(size_t)t * XP + DTR + threadIdx.x];
        __syncthreads();
        const float dtv = dt[(size_t)t * DI + ch];
        const float xv  = xc[(size_t)t * DI + ch];
        float acc = 0.f;
        #pragma unroll
        for (int n = 0; n < DS; ++n) {
            float dA = __expf(dtv * A[n]);
            h[n] = dA * h[n] + dtv * sBC[n] * xv;
            acc += h[n] * sBC[DS + n];
        }
        const int lsrc = dir ? (L - 1 - t) : t;
        const float zv = xz[(size_t)lsrc * XZW + DI + ch];
        y[(size_t)t * DI + ch] = (acc + dv * xv) * (zv / (1.f + __expf(-zv)));
        __syncthreads();
    }
}

// ---------------------------------------------------------------------------
// Host orchestration
// ---------------------------------------------------------------------------
static inline int gemm_blocks(int M, int N) {
    int tilesN = N >> 4;
    int groups = (M >> 6) * ((tilesN + 1) >> 1);
    return (groups + 7) / 8;
}

extern "C" void kernel_launch(void* const* d_in, const int* in_sizes, int n_in,
                              void* d_out, int out_size, void* d_ws, size_t ws_size,
                              hipStream_t stream) {
    const int*   ids      = (const int*)  d_in[0];
    const float* emb      = (const float*)d_in[1];
    const float* in_w     = (const float*)d_in[2];   // (2, 3072, 768)
    const float* conv_w   = (const float*)d_in[3];   // (2,2,1536,4)
    const float* conv_b   = (const float*)d_in[4];   // (2,2,1536)
    const float* xp_w     = (const float*)d_in[5];   // (2,2,80,1536)
    const float* dt_w     = (const float*)d_in[6];   // (2,2,1536,48)
    const float* dt_b     = (const float*)d_in[7];   // (2,2,1536)
    const float* a_log    = (const float*)d_in[8];   // (2,2,1536,16)
    const float* dp       = (const float*)d_in[9];   // (2,2,1536)
    const float* out_w    = (const float*)d_in[10];  // (2, 768, 1536)
    const float* norm_w   = (const float*)d_in[11];  // (2,768)
    const float* norm_b   = (const float*)d_in[12];
    const float* normf_w  = (const float*)d_in[13];  // (768)
    const float* normf_b  = (const float*)d_in[14];
    float* out = (float*)d_out;

    float* ws       = (float*)d_ws;
    float* hidden   = ws;                              // L*D
    float* residual = hidden   + (size_t)L * D;        // L*D
    float* hn       = residual + (size_t)L * D;        // L*D
    float* xz       = hn       + (size_t)L * D;        // L*XZW
    float* xc       = xz       + (size_t)L * XZW;      // L*DI
    float* dtb      = xc       + (size_t)L * DI;       // L*DI
    float* yb       = dtb      + (size_t)L * DI;       // L*DI
    float* dbl      = yb       + (size_t)L * DI;       // L*XP
    float* o0       = dbl      + (size_t)L * XP;       // L*D
    float* o1       = o0       + (size_t)L * D;        // L*D

    const int TPB = 256;
    const int nLD  = (L * D  + TPB - 1) / TPB;
    const int nLDI = (L * DI + TPB - 1) / TPB;

    embed_kernel<<<nLD, TPB, 0, stream>>>(ids, emb, hidden);
    hipMemcpyAsync(residual, hidden, (size_t)L * D * sizeof(float),
                   hipMemcpyDeviceToDevice, stream);

    for (int i = 0; i < 2; ++i) {
        if (i > 0)
            add_inplace_kernel<<<nLD, TPB, 0, stream>>>(residual, hidden, L * D);
        layernorm_kernel<<<L, TPB, 0, stream>>>(residual,
                                                norm_w + i * D, norm_b + i * D, hn);
        // xz = hn @ in_w[i]^T  (shared by both directions)
        gemm_f32_wmma<<<gemm_blocks(L, XZW), TPB, 0, stream>>>(
            hn, D, in_w + (size_t)i * XZW * D, xz, L, XZW, D);

        for (int dir = 0; dir < 2; ++dir) {
            const size_t s = (size_t)(i * 2 + dir);
            conv_silu_kernel<<<nLDI, TPB, 0, stream>>>(
                xz, conv_w + s * DI * 4, conv_b + s * DI, xc, dir);
            // dbl = xc @ xp_w^T
            gemm_f32_wmma<<<gemm_blocks(L, XP), TPB, 0, stream>>>(
                xc, DI, xp_w + s * XP * DI, dbl, L, XP, DI);
            // dt_raw = dbl[:, :48] @ dt_w^T   (strided A: lda = 80)
            gemm_f32_wmma<<<gemm_blocks(L, DI), TPB, 0, stream>>>(
                dbl, XP, dt_w + s * DI * DTR, dtb, L, DI, DTR);
            bias_softplus_kernel<<<nLDI, TPB, 0, stream>>>(dtb, dt_b + s * DI);
            scan_kernel<<<DI / TPB, TPB, 0, stream>>>(
                dtb, xc, dbl, a_log + s * DI * DS, dp + s * DI, xz, yb, dir);
            // o = yb @ out_w[i]^T
            gemm_f32_wmma<<<gemm_blocks(L, D), TPB, 0, stream>>>(
                yb, DI, out_w + (size_t)i * D * DI, dir ? o1 : o0, L, D, DI);
        }
        combine_kernel<<<nLD, TPB, 0, stream>>>(o0, o1, hidden);
    }
    add_inplace_kernel<<<nLD, TPB, 0, stream>>>(residual, hidden, L * D);
    layernorm_kernel<<<L, TPB, 0, stream>>>(residual, normf_w, normf_b, out);
}